// iVGAE_Encoder_7121055776880
// MI455X (gfx1250) — compile-verified
//
#include <hip/hip_runtime.h>
#include <hip/hip_bf16.h>
#include <stdint.h>

typedef __attribute__((ext_vector_type(2))) float v2f;
typedef __attribute__((ext_vector_type(8))) float v8f;

#define NODES_PER_TILE 16

// ---------------- utility kernels ----------------

__global__ void fill_f32(float* __restrict__ p, float v, long long n) {
    long long i = (long long)blockIdx.x * blockDim.x + threadIdx.x;
    if (i < n) p[i] = v;
}

__global__ void degree_kernel(const long long* __restrict__ dst, float* __restrict__ deg,
                              long long E) {
    long long e = (long long)blockIdx.x * blockDim.x + threadIdx.x;
    if (e < E) atomicAdd(&deg[dst[e]], 1.0f);
}

__global__ void rsqrt_kernel(float* __restrict__ d, long long n) {
    long long i = (long long)blockIdx.x * blockDim.x + threadIdx.x;
    if (i < n) {
        float v = d[i];
        d[i] = (v > 0.0f) ? rsqrtf(v) : 0.0f;
    }
}

// edge norm precompute: enorm[e] = dinv[src[e]] * dinv[dst[e]]  (reused by both layers)
__global__ void edge_norm_kernel(const long long* __restrict__ src,
                                 const long long* __restrict__ dst,
                                 const float* __restrict__ dinv,
                                 float* __restrict__ enorm, long long E) {
    long long e = (long long)blockIdx.x * blockDim.x + threadIdx.x;
    if (e < E) enorm[e] = dinv[src[e]] * dinv[dst[e]];
}

// ---------------- WMMA GEMM: C[M,N] = A[M,64] @ W[64,N] (+bias)(+relu) ----------------
// One wave per 16x16 output tile, full-precision v_wmma_f32_16x16x4_f32, K fully unrolled.
// blockDim.x = 32 * (N/16); blockIdx.x = row tile.

template <int N>
__global__ void gemm_wmma_f32(const float* __restrict__ A, const float* __restrict__ W,
                              const float* __restrict__ bias, float* __restrict__ C,
                              int M, int relu) {
    constexpr int K = 64;
    const int wave = threadIdx.x >> 5;
    const int lane = threadIdx.x & 31;
    const int half = lane >> 4;   // 0: lanes 0-15, 1: lanes 16-31
    const int l    = lane & 15;
    const int m0   = blockIdx.x * NODES_PER_TILE;
    const int n0   = wave * 16;

    // defensive per-lane row clamp (M = 100000 is an exact multiple of 16)
    const int rowA = (m0 + l < M) ? (m0 + l) : (M - 1);
    const float* __restrict__ arow = A + (size_t)rowA * K;
    const float* __restrict__ wcol = W + n0 + l;

    v8f acc = {};
#pragma unroll
    for (int k0 = 0; k0 < K; k0 += 4) {
        const int ka = k0 + 2 * half;            // this half-wave's K pair
        v2f a, b;
        a.x = arow[ka];
        a.y = arow[ka + 1];
        b.x = wcol[(size_t)ka * N];              // B row ka
        b.y = wcol[(size_t)(ka + 1) * N];        // B row ka+1
        acc = __builtin_amdgcn_wmma_f32_16x16x4_f32(
            /*neg_a=*/false, a, /*neg_b=*/false, b,
            /*c_mod=*/(short)0, acc, /*reuse_a=*/false, /*reuse_b=*/false);
    }

    const float bv = bias ? bias[n0 + l] : 0.0f;
#pragma unroll
    for (int r = 0; r < 8; ++r) {
        int row = m0 + r + 8 * half;             // D VGPR r -> row r (+8 for high half)
        if (row < M) {
            float v = acc[r] + bv;
            if (relu) v = fmaxf(v, 0.0f);
            C[(size_t)row * N + n0 + l] = v;
        }
    }
}

// ---------------- edge scatter: agg[dst] += lin[src] * enorm[e] ----------------
// 64 features; 16 threads per edge, float4 per thread.

__global__ void scatter_edges64(const float* __restrict__ lin,
                                const float* __restrict__ enorm,
                                const long long* __restrict__ src,
                                const long long* __restrict__ dst,
                                float* __restrict__ agg, long long E) {
    long long idx = (long long)blockIdx.x * blockDim.x + threadIdx.x;
    long long total = E << 4;                    // E * 16 groups of 4 feats
    if (idx >= total) return;
    long long e = idx >> 4;
    int g = (int)(idx & 15);
    long long s = src[e];
    long long d = dst[e];
    float nv = enorm[e];
    const float4 v = ((const float4*)(lin + s * 64))[g];
    float* o = agg + d * 64 + (long long)g * 4;
    atomicAdd(o + 0, v.x * nv);
    atomicAdd(o + 1, v.y * nv);
    atomicAdd(o + 2, v.z * nv);
    atomicAdd(o + 3, v.w * nv);
}

// ---------------- finalize: out = relu?(agg + dinv^2 * lin + b) ; in-place into agg ----------

__global__ void finalize64(float* __restrict__ agg, const float* __restrict__ lin,
                           const float* __restrict__ dinv, const float* __restrict__ bias,
                           long long nNodes, int relu) {
    long long idx = (long long)blockIdx.x * blockDim.x + threadIdx.x;
    long long total = nNodes * 64;
    if (idx >= total) return;
    long long i = idx >> 6;
    int c = (int)(idx & 63);
    float di = dinv[i];
    float v = agg[idx] + di * di * lin[idx] + bias[c];
    if (relu) v = fmaxf(v, 0.0f);
    agg[idx] = v;
}

// ---------------- host launcher ----------------

extern "C" void kernel_launch(void* const* d_in, const int* in_sizes, int n_in,
                              void* d_out, int out_size, void* d_ws, size_t ws_size,
                              hipStream_t stream) {
    const float*     x  = (const float*)d_in[0];
    const long long* ei = (const long long*)d_in[1];  // int64 edge_index [2, E]
    const float* W0 = (const float*)d_in[2];
    const float* b0 = (const float*)d_in[3];
    const float* W1 = (const float*)d_in[4];
    const float* b1 = (const float*)d_in[5];
    const float* Wm = (const float*)d_in[6];
    const float* bm = (const float*)d_in[7];
    const float* Ws = (const float*)d_in[8];
    const float* bs = (const float*)d_in[9];

    const long long Nn = in_sizes[0] / 64;     // 100000 nodes
    const long long E  = in_sizes[1] / 2;      // 1600000 edges
    const long long* src = ei;
    const long long* dst = ei + E;

    // workspace layout: dinv | enorm | B0 | B1   (feature buffers are Nn*64 f32)
    size_t feat = (size_t)Nn * 64;
    float* dinv  = (float*)d_ws;
    float* enorm = dinv + (((size_t)Nn + 255) & ~(size_t)255);
    float* B0    = enorm + (((size_t)E + 255) & ~(size_t)255);  // lin buffer
    float* B1    = B0 + feat;                                    // agg / h buffer

    const int TB = 256;
    dim3 b256(TB);
    auto blocks = [](long long n, int tb) { return dim3((unsigned)((n + tb - 1) / tb)); };

    // --- degrees, dinv, per-edge norms (shared by both conv layers) ---
    fill_f32<<<blocks(Nn, TB), b256, 0, stream>>>(dinv, 1.0f, Nn);   // self loops
    degree_kernel<<<blocks(E, TB), b256, 0, stream>>>(dst, dinv, E);
    rsqrt_kernel<<<blocks(Nn, TB), b256, 0, stream>>>(dinv, Nn);
    edge_norm_kernel<<<blocks(E, TB), b256, 0, stream>>>(src, dst, dinv, enorm, E);

    const unsigned mtiles = (unsigned)((Nn + NODES_PER_TILE - 1) / NODES_PER_TILE);
    dim3 gemmGrid(mtiles);
    dim3 gemmBlock64(32 * (64 / 16));   // 4 waves, N=64
    dim3 gemmBlock32(32 * (32 / 16));   // 2 waves, N=32

    // --- layer 0: lin0 = x @ W0 -> B0 ; agg in B1 ; h1 -> B1 (in place) ---
    gemm_wmma_f32<64><<<gemmGrid, gemmBlock64, 0, stream>>>(x, W0, nullptr, B0, (int)Nn, 0);
    fill_f32<<<blocks(feat, TB), b256, 0, stream>>>(B1, 0.0f, feat);
    scatter_edges64<<<blocks(E * 16, TB), b256, 0, stream>>>(B0, enorm, src, dst, B1, E);
    finalize64<<<blocks(feat, TB), b256, 0, stream>>>(B1, B0, dinv, b0, Nn, 1);

    // --- layer 1: lin1 = h1 @ W1 -> B0 ; agg in B1 (h1 consumed) ; h2 -> B1 ---
    gemm_wmma_f32<64><<<gemmGrid, gemmBlock64, 0, stream>>>(B1, W1, nullptr, B0, (int)Nn, 0);
    fill_f32<<<blocks(feat, TB), b256, 0, stream>>>(B1, 0.0f, feat);
    scatter_edges64<<<blocks(E * 16, TB), b256, 0, stream>>>(B0, enorm, src, dst, B1, E);
    finalize64<<<blocks(feat, TB), b256, 0, stream>>>(B1, B0, dinv, b1, Nn, 1);

    // --- heads: mean, logstd (concatenated flat in d_out) ---
    float* mean_out   = (float*)d_out;
    float* logstd_out = mean_out + (size_t)Nn * 32;
    gemm_wmma_f32<32><<<gemmGrid, gemmBlock32, 0, stream>>>(B1, Wm, bm, mean_out, (int)Nn, 0);
    gemm_wmma_f32<32><<<gemmGrid, gemmBlock32, 0, stream>>>(B1, Ws, bs, logstd_out, (int)Nn, 0);
}